// NonLocalBlock_79946521248296
// MI455X (gfx1250) — compile-verified
//
#include <hip/hip_runtime.h>

typedef __attribute__((ext_vector_type(2))) float v2f;
typedef __attribute__((ext_vector_type(8))) float v8f;

#define N_PIX 4096
#define CH 32
#define CIN 64
#define BATCH 4
#define PLANE (CH * N_PIX)  // 131072 floats per batch for Q/K/V/Y

// ---------------------------------------------------------------------------
// Kernel 1: fused Q/K/V 1x1 convs.  x:[B,64,N] natural; out Q/K/V:[B,32,N].
// One thread per (b,n); x column held in registers, weights are wave-uniform
// (scalar loads).  Bandwidth-bound: reads 4MB, writes 6MB.
// ---------------------------------------------------------------------------
__global__ __launch_bounds__(256) void qkv_conv_kernel(
    const float* __restrict__ x,
    const float* __restrict__ w1, const float* __restrict__ bb1,
    const float* __restrict__ w2, const float* __restrict__ bb2,
    const float* __restrict__ w3, const float* __restrict__ bb3,
    float* __restrict__ Q, float* __restrict__ K, float* __restrict__ V)
{
    int tid = blockIdx.x * blockDim.x + threadIdx.x;
    int b = tid >> 12;
    int n = tid & (N_PIX - 1);
    const float* xb = x + (size_t)b * CIN * N_PIX + n;
    float xr[CIN];
#pragma unroll
    for (int c = 0; c < CIN; ++c) xr[c] = xb[(size_t)c * N_PIX];
    size_t obase = (size_t)b * PLANE + n;
#pragma unroll 2
    for (int o = 0; o < CH; ++o) {
        float a1 = bb1[o], a2 = bb2[o], a3 = bb3[o];
#pragma unroll
        for (int c = 0; c < CIN; ++c) {
            float xv = xr[c];
            a1 = fmaf(w1[o * CIN + c], xv, a1);
            a2 = fmaf(w2[o * CIN + c], xv, a2);
            a3 = fmaf(w3[o * CIN + c], xv, a3);
        }
        Q[obase + (size_t)o * N_PIX] = a1;
        K[obase + (size_t)o * N_PIX] = a2;
        V[obase + (size_t)o * N_PIX] = a3;
    }
}

// ---------------------------------------------------------------------------
// Kernel 2: fused attention.
//   x1[b]  = Q[b] flat viewed as [4096,32] row-major  (raw-reshape trick)
//   x2[b]  = K[b] natural [32,4096]
//   v16[b] = V[b] flat viewed as [4096,32] row-major
//   S[b] = x1 @ x2  (16x16 tile per wave per j-step, WMMA f32 16x16x4, 8 ksteps)
//   P    = softmax over the BATCH axis -> 4-way pointwise, exchanged via LDS
//          between the block's 4 waves (wave w == batch w)
//   Y[b][i,:] += P @ v16   (accumulated in registers, 2 C-tiles of 16 cols)
// Never materializes the 268MB attention matrix.
// ---------------------------------------------------------------------------
__global__ __launch_bounds__(128) void attn_kernel(
    const float* __restrict__ Q, const float* __restrict__ K,
    const float* __restrict__ V, float* __restrict__ Y)
{
    __shared__ float sc[BATCH][16][17];  // raw scores, +1 pad vs bank conflicts
    __shared__ float pc[BATCH][16][17];  // normalized probabilities

    const int lane = threadIdx.x & 31;
    const int b    = threadIdx.x >> 5;   // wave id == batch id
    const int lo   = lane & 15;
    const int hi   = lane >> 4;          // half-wave select (K-pairs / row+8)
    const int i0   = blockIdx.x * 16;

    const float* Qb = Q + (size_t)b * PLANE;
    const float* Kb = K + (size_t)b * PLANE;
    const float* Vb = V + (size_t)b * PLANE;

    // Preload A-fragments of x1 rows i0..i0+15 (all 32 K, 8 ksteps of 4).
    // A 16x4 layout: lanes0-15 hold (M=lo, K=0,1), lanes16-31 (M=lo, K=2,3).
    v2f qa[8];
#pragma unroll
    for (int kk = 0; kk < 8; ++kk) {
        const float* p = Qb + (size_t)(i0 + lo) * CH + 4 * kk + 2 * hi;
        qa[kk] = v2f{p[0], p[1]};
    }

    v8f acc0 = {};  // y columns 0..15
    v8f acc1 = {};  // y columns 16..31

    for (int jt = 0; jt < N_PIX; jt += 16) {
        // ---- S tile = x1[i-tile] @ x2[:, j-tile] ----
        v8f s = {};
#pragma unroll
        for (int kk = 0; kk < 8; ++kk) {
            v2f kb;
            kb.x = Kb[(size_t)(4 * kk + 2 * hi)     * N_PIX + jt + lo];
            kb.y = Kb[(size_t)(4 * kk + 2 * hi + 1) * N_PIX + jt + lo];
            s = __builtin_amdgcn_wmma_f32_16x16x4_f32(
                    false, qa[kk], false, kb, (short)0, s, false, false);
        }
        // C layout -> row-major LDS: VGPR v, lanes0-15 row v, lanes16-31 row v+8
#pragma unroll
        for (int v = 0; v < 8; ++v) sc[b][v + 8 * hi][lo] = s[v];
        __syncthreads();

        // ---- softmax across the 4 batches (pointwise per (i,j)) ----
#pragma unroll
        for (int v = 0; v < 8; ++v) {
            int r = v + 8 * hi;
            float s0 = sc[0][r][lo], s1 = sc[1][r][lo];
            float s2 = sc[2][r][lo], s3 = sc[3][r][lo];
            float m = fmaxf(fmaxf(s0, s1), fmaxf(s2, s3));
            float e0 = __expf(s0 - m), e1 = __expf(s1 - m);
            float e2 = __expf(s2 - m), e3 = __expf(s3 - m);
            float d = e0 + e1 + e2 + e3;
            float eb = (b == 0) ? e0 : (b == 1) ? e1 : (b == 2) ? e2 : e3;
            pc[b][r][lo] = eb / d;
        }
        __syncthreads();

        // ---- accumulate P (16x16) @ V-tile (16x32): 4 ksteps x 2 col-halves
#pragma unroll
        for (int kk = 0; kk < 4; ++kk) {
            v2f a;
            a.x = pc[b][lo][4 * kk + 2 * hi];
            a.y = pc[b][lo][4 * kk + 2 * hi + 1];
            const float* vr0 = Vb + (size_t)(jt + 4 * kk + 2 * hi) * CH;
            const float* vr1 = vr0 + CH;
            v2f bl, bh;
            bl.x = vr0[lo];      bl.y = vr1[lo];
            bh.x = vr0[16 + lo]; bh.y = vr1[16 + lo];
            acc0 = __builtin_amdgcn_wmma_f32_16x16x4_f32(
                       false, a, false, bl, (short)0, acc0, false, false);
            acc1 = __builtin_amdgcn_wmma_f32_16x16x4_f32(
                       false, a, false, bh, (short)0, acc1, false, false);
        }
    }

    // ---- store Y flat as [4096][32] row-major per batch ----
    float* Yb = Y + (size_t)b * PLANE;
#pragma unroll
    for (int v = 0; v < 8; ++v) {
        int r = v + 8 * hi;
        Yb[(size_t)(i0 + r) * CH + lo]      = acc0[v];
        Yb[(size_t)(i0 + r) * CH + 16 + lo] = acc1[v];
    }
}

// ---------------------------------------------------------------------------
// Kernel 3: final 1x1 conv.  Y flat reinterpreted as [B,32,N]; out:[B,64,N].
// ---------------------------------------------------------------------------
__global__ __launch_bounds__(256) void out_conv_kernel(
    const float* __restrict__ Y, const float* __restrict__ w4,
    const float* __restrict__ bb4, float* __restrict__ out)
{
    int tid = blockIdx.x * blockDim.x + threadIdx.x;
    int b = tid >> 12;
    int n = tid & (N_PIX - 1);
    const float* Yb = Y + (size_t)b * PLANE;  // viewed as [32][4096]
    float yr[CH];
#pragma unroll
    for (int c = 0; c < CH; ++c) yr[c] = Yb[(size_t)c * N_PIX + n];
    float* ob = out + (size_t)b * CIN * N_PIX + n;
#pragma unroll 2
    for (int o = 0; o < CIN; ++o) {
        float a = bb4[o];
#pragma unroll
        for (int c = 0; c < CH; ++c) a = fmaf(w4[o * CH + c], yr[c], a);
        ob[(size_t)o * N_PIX] = a;
    }
}

// ---------------------------------------------------------------------------
extern "C" void kernel_launch(void* const* d_in, const int* in_sizes, int n_in,
                              void* d_out, int out_size, void* d_ws, size_t ws_size,
                              hipStream_t stream) {
    const float* x   = (const float*)d_in[0];
    const float* w1  = (const float*)d_in[1];
    const float* bb1 = (const float*)d_in[2];
    const float* w2  = (const float*)d_in[3];
    const float* bb2 = (const float*)d_in[4];
    const float* w3  = (const float*)d_in[5];
    const float* bb3 = (const float*)d_in[6];
    const float* w4  = (const float*)d_in[7];
    const float* bb4 = (const float*)d_in[8];
    float* out = (float*)d_out;

    // Workspace layout: Q | K | V | Y, each BATCH*PLANE floats (2MB) = 8MB total
    float* Q = (float*)d_ws;
    float* K = Q + (size_t)BATCH * PLANE;
    float* V = K + (size_t)BATCH * PLANE;
    float* Y = V + (size_t)BATCH * PLANE;

    qkv_conv_kernel<<<(BATCH * N_PIX) / 256, 256, 0, stream>>>(
        x, w1, bb1, w2, bb2, w3, bb3, Q, K, V);
    attn_kernel<<<N_PIX / 16, 128, 0, stream>>>(Q, K, V, Y);
    out_conv_kernel<<<(BATCH * N_PIX) / 256, 256, 0, stream>>>(Y, w4, bb4, out);
}